// GAT_12506944766356
// MI455X (gfx1250) — compile-verified
//
#include <hip/hip_runtime.h>
#include <math.h>

#define N_NODES 50000
#define N_EDGES 800000
#define HEADS   4
#define DIM     32
#define F       128          // HEADS*DIM == in_dim
#define SLOPE   0.2f

typedef __attribute__((ext_vector_type(2))) float v2f;
typedef __attribute__((ext_vector_type(8))) float v8f;

// ---------------------------------------------------------------------------
// feat[N,128] = h[N,128] @ W[128,128]   (row-major, fp32, WMMA 16x16x4 f32)
// Block = 256 threads (8 waves). Block computes a 16-row x 128-col panel.
// Wave w owns columns [16w, 16w+16). K looped 128/4 = 32 WMMA steps.
// N_NODES = 3125 * 16 exactly -> no guards, EXEC stays all-ones (WMMA req).
// ---------------------------------------------------------------------------
__global__ __launch_bounds__(256)
void gat_gemm_wmma(const float* __restrict__ hmat,
                   const float* __restrict__ W,
                   float* __restrict__ feat)
{
    __shared__ float As[16 * 132];          // 16 rows, padded stride 132

    const int m0 = blockIdx.x * 16;

    // Stage the 16x128 A-panel: each thread moves 8 consecutive floats.
    {
        const int row  = threadIdx.x >> 4;          // 0..15
        const int col0 = (threadIdx.x & 15) * 8;    // 0..120
        const float* s = hmat + (size_t)(m0 + row) * F + col0;
        float4 p0 = *(const float4*)(s);
        float4 p1 = *(const float4*)(s + 4);
        float* d = As + row * 132 + col0;
        *(float4*)(d)     = p0;
        *(float4*)(d + 4) = p1;
    }
    __syncthreads();

    const int wave    = threadIdx.x >> 5;   // 0..7
    const int lane    = threadIdx.x & 31;
    const int halfsel = lane >> 4;          // 0: lanes 0-15, 1: lanes 16-31
    const int l16     = lane & 15;
    const int n0      = wave * 16;

    // Preload this wave's 128x16 B-tile of W into registers (64 VGPRs).
    // ISA B layout (4x16, 2 VGPRs): VGPR0 = rows {k0, k0+2}, VGPR1 = {k0+1, k0+3}.
    float bx[32], by[32];
#pragma unroll
    for (int ks = 0; ks < 32; ++ks) {
        const int k0 = ks * 4 + halfsel * 2;
        bx[ks] = W[(size_t)(k0    ) * F + n0 + l16];
        by[ks] = W[(size_t)(k0 + 1) * F + n0 + l16];
    }

    v8f c = {};
#pragma unroll
    for (int ks = 0; ks < 32; ++ks) {
        const int k0 = ks * 4 + halfsel * 2;
        v2f a;                               // A 16x4: lane<16 -> K={4k,4k+1}, lane>=16 -> K={4k+2,4k+3}
        a.x = As[l16 * 132 + k0];
        a.y = As[l16 * 132 + k0 + 1];
        v2f b;
        b.x = bx[ks];
        b.y = by[ks];
        c = __builtin_amdgcn_wmma_f32_16x16x4_f32(
                /*neg_a=*/false, a, /*neg_b=*/false, b,
                /*c_mod=*/(short)0, c, /*reuse_a=*/false, /*reuse_b=*/false);
    }

    // C/D layout: VGPR r -> (M = r + 8*halfsel, N = l16)
#pragma unroll
    for (int r = 0; r < 8; ++r) {
        const int mm = m0 + r + halfsel * 8;
        feat[(size_t)mm * F + n0 + l16] = c[r];
    }
}

// ---------------------------------------------------------------------------
// el[n,h] = sum_d feat[n,h,d]*al[h,d] ; er likewise with ar.
// ---------------------------------------------------------------------------
__global__ void gat_attn(const float* __restrict__ feat,
                         const float* __restrict__ al,
                         const float* __restrict__ ar,
                         float* __restrict__ el,
                         float* __restrict__ er)
{
    const int idx = blockIdx.x * blockDim.x + threadIdx.x;
    if (idx >= N_NODES * HEADS) return;
    const int hh = idx & (HEADS - 1);
    const int n  = idx >> 2;
    const float* f = feat + (size_t)n * F + hh * DIM;
    float sl = 0.f, sr = 0.f;
#pragma unroll
    for (int d = 0; d < DIM; ++d) {
        const float v = f[d];
        sl += v * al[hh * DIM + d];
        sr += v * ar[hh * DIM + d];
    }
    el[idx] = sl;
    er[idx] = sr;
}

// rst = h + bias  (folds the identity residual and the bias add)
__global__ void gat_init_rst(const float* __restrict__ hcur,
                             const float* __restrict__ bias,
                             float* __restrict__ rst)
{
    const int idx = blockIdx.x * blockDim.x + threadIdx.x;
    if (idx >= N_NODES * F) return;
    rst[idx] = hcur[idx] + bias[idx & (F - 1)];
}

__global__ void gat_init_softmax(float* __restrict__ emax,
                                 float* __restrict__ denom)
{
    const int idx = blockIdx.x * blockDim.x + threadIdx.x;
    if (idx >= N_NODES * HEADS) return;
    emax[idx]  = -INFINITY;
    denom[idx] = 0.f;
}

__device__ __forceinline__ float lrelu(float x) {
    return x > 0.f ? x : SLOPE * x;
}

// float atomic max via monotone int/uint reinterpretation (native L2 atomics)
__device__ __forceinline__ void atomicMaxF(float* addr, float v) {
    if (v >= 0.f) atomicMax((int*)addr, __float_as_int(v));
    else          atomicMin((unsigned int*)addr, __float_as_uint(v));
}

__global__ void gat_edge_max(const int* __restrict__ src,
                             const int* __restrict__ dst,
                             const float* __restrict__ el,
                             const float* __restrict__ er,
                             float* __restrict__ emax)
{
    const int idx = blockIdx.x * blockDim.x + threadIdx.x;
    if (idx >= N_EDGES * HEADS) return;
    const int hh = idx & (HEADS - 1);
    const int ed = idx >> 2;
    const int s = src[ed], t = dst[ed];
    const float e = lrelu(el[s * HEADS + hh] + er[t * HEADS + hh]);
    atomicMaxF(&emax[t * HEADS + hh], e);
}

__global__ void gat_edge_exp(const int* __restrict__ src,
                             const int* __restrict__ dst,
                             const float* __restrict__ el,
                             const float* __restrict__ er,
                             const float* __restrict__ emax,
                             float* __restrict__ ee,
                             float* __restrict__ denom)
{
    const int idx = blockIdx.x * blockDim.x + threadIdx.x;
    if (idx >= N_EDGES * HEADS) return;
    const int hh = idx & (HEADS - 1);
    const int ed = idx >> 2;
    const int s = src[ed], t = dst[ed];
    const float e = lrelu(el[s * HEADS + hh] + er[t * HEADS + hh]);
    const float x = __expf(e - emax[t * HEADS + hh]);
    ee[idx] = x;                       // idx == ed*HEADS + hh
    atomicAdd(&denom[t * HEADS + hh], x);
}

// 32 lanes per edge; lane j handles feature columns [4j, 4j+4) (head j/8).
__global__ void gat_edge_agg(const int* __restrict__ src,
                             const int* __restrict__ dst,
                             const float* __restrict__ feat,
                             const float* __restrict__ ee,
                             const float* __restrict__ denom,
                             float* __restrict__ rst)
{
    const long long idx = (long long)blockIdx.x * blockDim.x + threadIdx.x;
    if (idx >= (long long)N_EDGES * 32) return;
    const int j  = (int)(idx & 31);
    const int ed = (int)(idx >> 5);
    const int s = src[ed], t = dst[ed];
    const int hh = j >> 3;
    const float alpha = ee[ed * HEADS + hh] / denom[t * HEADS + hh];
    const float4 f = *(const float4*)(feat + (size_t)s * F + j * 4);
    float* o = rst + (size_t)t * F + j * 4;
    atomicAdd(o + 0, f.x * alpha);
    atomicAdd(o + 1, f.y * alpha);
    atomicAdd(o + 2, f.z * alpha);
    atomicAdd(o + 3, f.w * alpha);
}

// out[n,d] = mean over heads of rst[n,h,d]
__global__ void gat_final_mean(const float* __restrict__ rst,
                               float* __restrict__ out)
{
    const int idx = blockIdx.x * blockDim.x + threadIdx.x;
    if (idx >= N_NODES * DIM) return;
    const int n = idx >> 5;
    const int d = idx & 31;
    const float* r = rst + (size_t)n * F + d;
    out[idx] = 0.25f * (r[0] + r[DIM] + r[2 * DIM] + r[3 * DIM]);
}

// ---------------------------------------------------------------------------
extern "C" void kernel_launch(void* const* d_in, const int* in_sizes, int n_in,
                              void* d_out, int out_size, void* d_ws, size_t ws_size,
                              hipStream_t stream)
{
    const float* h   = (const float*)d_in[0];
    const int*   src = (const int*)d_in[1];
    const int*   dst = (const int*)d_in[2];
    const float* Wl[3]  = {(const float*)d_in[3], (const float*)d_in[7],  (const float*)d_in[11]};
    const float* all[3] = {(const float*)d_in[4], (const float*)d_in[8],  (const float*)d_in[12]};
    const float* arl[3] = {(const float*)d_in[5], (const float*)d_in[9],  (const float*)d_in[13]};
    const float* bl[3]  = {(const float*)d_in[6], (const float*)d_in[10], (const float*)d_in[14]};

    float* ws = (float*)d_ws;
    const size_t NF = (size_t)N_NODES * F;
    const size_t NH = (size_t)N_NODES * HEADS;
    float* bufA  = ws;                 // layer0 out, layer2 out
    float* bufB  = bufA  + NF;         // layer1 out
    float* feat  = bufB  + NF;
    float* el    = feat  + NF;
    float* er    = el    + NH;
    float* emax  = er    + NH;
    float* denom = emax  + NH;
    float* ee    = denom + NH;         // E*HEADS floats

    const int T = 256;
    const int gNH  = (int)((NH + T - 1) / T);
    const int gNF  = (int)((NF + T - 1) / T);
    const int gEH  = (N_EDGES * HEADS + T - 1) / T;
    const int gE32 = (int)(((long long)N_EDGES * 32 + T - 1) / T);
    const int gND  = (N_NODES * DIM + T - 1) / T;

    const float* hcur = h;
    float* rst = bufA;
    for (int l = 0; l < 3; ++l) {
        rst = (l == 1) ? bufB : bufA;

        gat_gemm_wmma<<<N_NODES / 16, 256, 0, stream>>>(hcur, Wl[l], feat);
        gat_attn<<<gNH, T, 0, stream>>>(feat, all[l], arl[l], el, er);
        gat_init_rst<<<gNF, T, 0, stream>>>(hcur, bl[l], rst);
        gat_init_softmax<<<gNH, T, 0, stream>>>(emax, denom);
        gat_edge_max<<<gEH, T, 0, stream>>>(src, dst, el, er, emax);
        gat_edge_exp<<<gEH, T, 0, stream>>>(src, dst, el, er, emax, ee, denom);
        gat_edge_agg<<<gE32, T, 0, stream>>>(src, dst, feat, ee, denom, rst);

        hcur = rst;
    }

    gat_final_mean<<<gND, T, 0, stream>>>(bufA, (float*)d_out);
}